// SetAbstract_29076928594033
// MI455X (gfx1250) — compile-verified
//
#include <hip/hip_runtime.h>
#include <hip/hip_bf16.h>

typedef __attribute__((ext_vector_type(2))) float v2f;
typedef __attribute__((ext_vector_type(8))) float v8f;

#define B_    16
#define N_    4096
#define NG    1024
#define NS    32
#define P_TOT (B_ * NG * NS)   // 524288
#define BN_EPS 1e-5f

// ---------------------------------------------------------------------------
// 1) Farthest point sampling: one block per batch, dist[] lives in LDS,
//    xyz served from L2. Serial 1024-step chain with block-wide argmax.
// ---------------------------------------------------------------------------
__global__ __launch_bounds__(256) void fps_kernel(const float* __restrict__ xyz,
                                                  int* __restrict__ fps_idx,
                                                  float* __restrict__ new_xyz) {
    __shared__ float dist[N_];
    __shared__ float rv[256];
    __shared__ int   ri[256];
    __shared__ float cen[3];
    __shared__ int   cur;
    const int b = blockIdx.x;
    const int t = threadIdx.x;
    const float* x = xyz + (size_t)b * N_ * 3;
    for (int i = t; i < N_; i += 256) dist[i] = 1e10f;
    if (t == 0) cur = 0;
    __syncthreads();
    for (int it = 0; it < NG; ++it) {
        if (t == 0) {
            int far = cur;
            fps_idx[b * NG + it] = far;
            float cx = x[far * 3 + 0], cy = x[far * 3 + 1], cz = x[far * 3 + 2];
            cen[0] = cx; cen[1] = cy; cen[2] = cz;
            float* o = new_xyz + ((size_t)b * NG + it) * 3;
            o[0] = cx; o[1] = cy; o[2] = cz;
        }
        __syncthreads();
        const float cx = cen[0], cy = cen[1], cz = cen[2];
        float bv = -1.0f; int bi = 0;
        #pragma unroll
        for (int k = 0; k < N_ / 256; ++k) {
            const int i = t + k * 256;
            float dx = x[i * 3 + 0] - cx, dy = x[i * 3 + 1] - cy, dz = x[i * 3 + 2] - cz;
            float d = dx * dx + dy * dy + dz * dz;
            d = fminf(dist[i], d);
            dist[i] = d;
            if (d > bv) { bv = d; bi = i; }
        }
        rv[t] = bv; ri[t] = bi;
        __syncthreads();
        for (int s = 128; s > 0; s >>= 1) {
            if (t < s) {
                float ov = rv[t + s]; int oi = ri[t + s];
                if (ov > rv[t] || (ov == rv[t] && oi < ri[t])) { rv[t] = ov; ri[t] = oi; }
            }
            __syncthreads();
        }
        if (t == 0) cur = ri[0];
        __syncthreads();
    }
}

// ---------------------------------------------------------------------------
// 2) Ball query: one wave32 per group; order-preserving first-32 selection
//    via ballot + popcount prefix (no sort needed).
// ---------------------------------------------------------------------------
__global__ __launch_bounds__(256) void ballq_kernel(const float* __restrict__ xyz,
                                                    const float* __restrict__ new_xyz,
                                                    int* __restrict__ idx_out) {
    const int lane = threadIdx.x & 31;
    const int wave = threadIdx.x >> 5;
    const int g = blockIdx.x * 8 + wave;          // group in [0, B*NG)
    const int b = g >> 10;
    const float r2 = 0.2f * 0.2f;
    const float* x = xyz + (size_t)b * N_ * 3;
    const float* c = new_xyz + (size_t)g * 3;
    const float cx = c[0], cy = c[1], cz = c[2];
    int* out = idx_out + (size_t)g * NS;
    int count = 0, first = -1;
    for (int base = 0; base < N_ && count < NS; base += 32) {
        const int j = base + lane;
        float dx = x[j * 3 + 0] - cx, dy = x[j * 3 + 1] - cy, dz = x[j * 3 + 2] - cz;
        const bool in = (dx * dx + dy * dy + dz * dz) <= r2;
        const unsigned mask = (unsigned)__ballot(in);
        if (first < 0 && mask) first = base + (__ffs(mask) - 1);
        const int pos = count + __popc(mask & ((1u << lane) - 1u));
        if (in && pos < NS) out[pos] = j;
        count += __popc(mask);
    }
    if (count < NS && lane >= count) out[lane] = first;   // pad with first hit
}

// ---------------------------------------------------------------------------
// 3) Layer 0: gather + concat + 6->64 linear (+bias), accumulate BN stats.
// ---------------------------------------------------------------------------
__global__ __launch_bounds__(256) void layer0_kernel(const float* __restrict__ xyz,
                                                     const float* __restrict__ feat,
                                                     const float* __restrict__ new_xyz,
                                                     const int* __restrict__ idx,
                                                     const float* __restrict__ w0,
                                                     const float* __restrict__ b0,
                                                     float* __restrict__ y0,
                                                     float* __restrict__ gsum,
                                                     float* __restrict__ gsq) {
    __shared__ float lsum[64], lsq[64];
    const int t = threadIdx.x;
    const int lane = t & 31;
    if (t < 64) { lsum[t] = 0.f; lsq[t] = 0.f; }
    __syncthreads();
    const int p = blockIdx.x * 256 + t;
    const int b = p >> 15;
    const int s = (p >> 5) & (NG - 1);
    const int j = idx[p];
    const float* xb = xyz + (size_t)b * N_ * 3;
    const float* fb = feat + (size_t)b * N_ * 3;
    const float* c = new_xyz + ((size_t)(b * NG + s)) * 3;
    float x6[6];
    x6[0] = xb[j * 3 + 0] - c[0]; x6[1] = xb[j * 3 + 1] - c[1]; x6[2] = xb[j * 3 + 2] - c[2];
    x6[3] = fb[j * 3 + 0]; x6[4] = fb[j * 3 + 1]; x6[5] = fb[j * 3 + 2];
    float* yo = y0 + (size_t)p * 64;
    for (int o = 0; o < 64; ++o) {
        float a = b0[o];
        #pragma unroll
        for (int cc = 0; cc < 6; ++cc) a += w0[o * 6 + cc] * x6[cc];
        yo[o] = a;
        float sv = a, qv = a * a;
        #pragma unroll
        for (int off = 16; off > 0; off >>= 1) {
            sv += __shfl_down(sv, off, 32);
            qv += __shfl_down(qv, off, 32);
        }
        if (lane == 0) { atomicAdd(&lsum[o], sv); atomicAdd(&lsq[o], qv); }
    }
    __syncthreads();
    if (t < 64) { atomicAdd(&gsum[t], lsum[t]); atomicAdd(&gsq[t], lsq[t]); }
}

// ---------------------------------------------------------------------------
// 4) Fold BN stats into per-channel scale/shift.
// ---------------------------------------------------------------------------
__global__ void finalize_kernel(const float* __restrict__ gsum, const float* __restrict__ gsq,
                                const float* __restrict__ g, const float* __restrict__ be,
                                float* __restrict__ scale, float* __restrict__ shift, int C) {
    const int c = threadIdx.x;
    if (c < C) {
        const float inv_n = 1.0f / (float)P_TOT;
        const float mu = gsum[c] * inv_n;
        const float var = gsq[c] * inv_n - mu * mu;
        const float sc = g[c] * rsqrtf(var + BN_EPS);
        scale[c] = sc;
        shift[c] = be[c] - mu * sc;
    }
}

// ---------------------------------------------------------------------------
// 5) WMMA GEMM layer: BN+ReLU fused into LDS staging, f32 16x16x4 WMMA,
//    optional y store / BN-stats / K-max (templated).
//    Fragment layout (ISA 7.12.2): A lane l -> row M=l&15, K=2*(l>=16)+{0,1};
//    B mirrors with N=l&15; C/D lane l VGPR i -> (M=i+8*(l>=16), N=l&15).
// ---------------------------------------------------------------------------
template <int WAVES, int NTILES, bool STORE_Y, bool DO_STATS, bool DO_MAX>
__global__ __launch_bounds__(WAVES * 32) void gemm_layer_kernel(
    const float* __restrict__ yin,        // [P, 64] pre-BN input
    const float* __restrict__ in_scale, const float* __restrict__ in_shift,
    const float* __restrict__ w,          // [NTILES*16, 64]
    const float* __restrict__ bias,       // [NTILES*16]
    float* __restrict__ yout,             // [P, NTILES*16]
    float* __restrict__ gsum, float* __restrict__ gsq,
    const float* __restrict__ out_scale, const float* __restrict__ out_shift,
    float* __restrict__ feat_out)         // [B*NG, NTILES*16]
{
    constexpr int PTS = WAVES * 16;
    constexpr int OC  = NTILES * 16;
    __shared__ float lx[PTS * 65];        // padded stride: conflict-free
    __shared__ float lw[OC * 65];
    __shared__ float lsum[OC], lsq[OC];
    const int t = threadIdx.x;
    const int lane = t & 31;
    const int waveid = t >> 5;
    const int half = lane >> 4;
    const int row = lane & 15;
    const size_t pbase = (size_t)blockIdx.x * PTS;

    // stage input tile, applying previous layer's BN + ReLU
    for (int f = t; f < PTS * 64; f += WAVES * 32) {
        const int pl = f >> 6, c = f & 63;
        float v = yin[(pbase + pl) * 64 + c];
        v = v * in_scale[c] + in_shift[c];
        lx[pl * 65 + c] = fmaxf(v, 0.0f);
    }
    // stage weights
    for (int f = t; f < OC * 64; f += WAVES * 32) {
        const int o = f >> 6, c = f & 63;
        lw[o * 65 + c] = w[f];
    }
    if constexpr (DO_STATS) {
        for (int c = t; c < OC; c += WAVES * 32) { lsum[c] = 0.f; lsq[c] = 0.f; }
    }
    __syncthreads();

    v8f acc[NTILES];
    #pragma unroll
    for (int tt = 0; tt < NTILES; ++tt) {
        const float bo = bias[tt * 16 + row];
        #pragma unroll
        for (int i = 0; i < 8; ++i) acc[tt][i] = bo;
    }

    const int prow = waveid * 16 + row;
    #pragma unroll
    for (int ks = 0; ks < 16; ++ks) {         // K = 64 in steps of 4
        const int kb = ks * 4 + 2 * half;
        v2f a; a[0] = lx[prow * 65 + kb]; a[1] = lx[prow * 65 + kb + 1];
        #pragma unroll
        for (int tt = 0; tt < NTILES; ++tt) {
            const int oc = tt * 16 + row;
            v2f bm; bm[0] = lw[oc * 65 + kb]; bm[1] = lw[oc * 65 + kb + 1];
            acc[tt] = __builtin_amdgcn_wmma_f32_16x16x4_f32(
                false, a, false, bm, (short)0, acc[tt], false, false);
        }
    }

    const size_t gp = pbase + (size_t)waveid * 16;
    #pragma unroll
    for (int tt = 0; tt < NTILES; ++tt) {
        const int oc = tt * 16 + row;
        if constexpr (STORE_Y) {
            #pragma unroll
            for (int i = 0; i < 8; ++i)
                yout[(gp + i + 8 * half) * OC + oc] = acc[tt][i];
        }
        if constexpr (DO_STATS) {
            float sv = 0.f, qv = 0.f;
            #pragma unroll
            for (int i = 0; i < 8; ++i) { const float v = acc[tt][i]; sv += v; qv += v * v; }
            atomicAdd(&lsum[oc], sv);
            atomicAdd(&lsq[oc], qv);
        }
        if constexpr (DO_MAX) {
            const float sc = out_scale[oc], sh = out_shift[oc];
            float m = 0.0f;                              // ReLU floor
            #pragma unroll
            for (int i = 0; i < 8; ++i) m = fmaxf(m, fmaxf(acc[tt][i] * sc + sh, 0.0f));
            m = fmaxf(m, __shfl_xor(m, 16, 32));         // combine row halves
            if (half == 0) {
                const size_t grp = gp >> 5;              // 32 samples per group
                atomicMax((int*)&feat_out[grp * OC + oc], __float_as_int(m));
            }
        }
    }
    if constexpr (DO_STATS) {
        __syncthreads();
        for (int c = t; c < OC; c += WAVES * 32) {
            atomicAdd(&gsum[c], lsum[c]);
            atomicAdd(&gsq[c], lsq[c]);
        }
    }
}

// ---------------------------------------------------------------------------
extern "C" void kernel_launch(void* const* d_in, const int* in_sizes, int n_in,
                              void* d_out, int out_size, void* d_ws, size_t ws_size,
                              hipStream_t stream) {
    const float* xyz  = (const float*)d_in[0];
    const float* feat = (const float*)d_in[1];
    const float* w0 = (const float*)d_in[2];  const float* b0 = (const float*)d_in[3];
    const float* g0 = (const float*)d_in[4];  const float* be0 = (const float*)d_in[5];
    const float* w1 = (const float*)d_in[6];  const float* b1 = (const float*)d_in[7];
    const float* g1 = (const float*)d_in[8];  const float* be1 = (const float*)d_in[9];
    const float* w2 = (const float*)d_in[10]; const float* b2 = (const float*)d_in[11];
    const float* g2 = (const float*)d_in[12]; const float* be2 = (const float*)d_in[13];

    float* out_xyz  = (float*)d_out;                 // [B, NG, 3]
    float* out_feat = out_xyz + (size_t)B_ * NG * 3; // [B, NG, 128]

    char* ws = (char*)d_ws;
    int*   fps_idx = (int*)ws;                               // 64 KB
    int*   bidx    = (int*)(ws + (64 << 10));                // 2 MB
    float* stats   = (float*)(ws + (64 << 10) + (2 << 20));  // sums + coeffs
    float* sum0 = stats;       float* sq0 = stats + 64;
    float* sum1 = stats + 128; float* sq1 = stats + 192;
    float* sum2 = stats + 256; float* sq2 = stats + 384;     // 128 each
    float* sc0 = stats + 512;  float* sh0 = sc0 + 64;
    float* sc1 = sh0 + 64;     float* sh1 = sc1 + 64;
    float* sc2 = sh1 + 64;     float* sh2 = sc2 + 128;
    float* y0 = (float*)(ws + (4 << 20));                    // [P,64] 134 MB
    float* y1 = y0 + (size_t)P_TOT * 64;                     // [P,64] 134 MB

    hipMemsetAsync(stats, 0, 512 * sizeof(float), stream);
    hipMemsetAsync(out_feat, 0, (size_t)B_ * NG * 128 * sizeof(float), stream);

    fps_kernel<<<B_, 256, 0, stream>>>(xyz, fps_idx, out_xyz);
    ballq_kernel<<<(B_ * NG) / 8, 256, 0, stream>>>(xyz, out_xyz, bidx);
    layer0_kernel<<<P_TOT / 256, 256, 0, stream>>>(xyz, feat, out_xyz, bidx,
                                                   w0, b0, y0, sum0, sq0);
    finalize_kernel<<<1, 64, 0, stream>>>(sum0, sq0, g0, be0, sc0, sh0, 64);
    gemm_layer_kernel<8, 4, true, true, false><<<P_TOT / 128, 256, 0, stream>>>(
        y0, sc0, sh0, w1, b1, y1, sum1, sq1, nullptr, nullptr, nullptr);
    finalize_kernel<<<1, 64, 0, stream>>>(sum1, sq1, g1, be1, sc1, sh1, 64);
    gemm_layer_kernel<4, 8, false, true, false><<<P_TOT / 64, 128, 0, stream>>>(
        y1, sc1, sh1, w2, b2, nullptr, sum2, sq2, nullptr, nullptr, nullptr);
    finalize_kernel<<<1, 128, 0, stream>>>(sum2, sq2, g2, be2, sc2, sh2, 128);
    gemm_layer_kernel<4, 8, false, false, true><<<P_TOT / 64, 128, 0, stream>>>(
        y1, sc1, sh1, w2, b2, nullptr, nullptr, nullptr, sc2, sh2, out_feat);
}